// GraphSAGENeighbor_28707561407282
// MI455X (gfx1250) — compile-verified
//
#include <hip/hip_runtime.h>

// ---------------------------------------------------------------------------
// GraphSAGE 2-layer (mean aggregation), fp32, for MI455X / gfx1250.
//   N0=1228800 src nodes (D_IN=128), N1=81920 mid nodes (D_H=256),
//   N2=8192 out nodes (D_OUT=64).
// Structure:
//   1) zero workspace accumulators
//   2) scatter0: agg0[dst0] += x[src0] (float4 gather + f32 global atomics -> L2)
//   3) layer0 GEMM: h1 = relu(x[:N1] @ Wself0 + (agg0/deg0) @ Wneigh0 + b0)
//      via V_WMMA_F32_16X16X4_F32, one wave per 16x16 tile
//   4) scatter1: agg1[dst1] += h1[src1]
//   5) layer1 GEMM: out = h1[:N2] @ Wself1 + (agg1/deg1) @ Wneigh1 + b1
// ---------------------------------------------------------------------------

typedef float v2f __attribute__((ext_vector_type(2)));
typedef float v8f __attribute__((ext_vector_type(8)));

#define CN0 1228800
#define CN1 81920
#define CN2 8192
#define CD_IN 128
#define CD_H 256
#define CD_OUT 64

// -------------------- workspace zeroing --------------------
__global__ void sage_zero_kernel(float* __restrict__ p, unsigned long long n) {
  unsigned long long i = (unsigned long long)blockIdx.x * blockDim.x + threadIdx.x;
  unsigned long long stride = (unsigned long long)gridDim.x * blockDim.x;
  for (; i < n; i += stride) p[i] = 0.0f;
}

// -------------------- layer-0 scatter (D=128, 32 chunks of 4 floats) -------
__global__ void sage_scatter0_kernel(const float* __restrict__ x,
                                     const int* __restrict__ src,
                                     const int* __restrict__ dst,
                                     float* __restrict__ agg,
                                     float* __restrict__ deg) {
  unsigned tid = blockIdx.x * blockDim.x + threadIdx.x;
  unsigned e = tid >> 5;               // edge index (128/4 = 32 chunks per edge)
  if (e >= CN0) return;
  unsigned c = (tid & 31u) << 2;       // element offset within row
  int s = src[e];
  int d = dst[e];
  const float4 v = *reinterpret_cast<const float4*>(x + (size_t)s * CD_IN + c);
  float* a = agg + (size_t)d * CD_IN + c;
  atomicAdd(a + 0, v.x);
  atomicAdd(a + 1, v.y);
  atomicAdd(a + 2, v.z);
  atomicAdd(a + 3, v.w);
  if ((tid & 31u) == 0u) atomicAdd(deg + d, 1.0f);
}

// -------------------- layer-1 scatter (D=256, 64 chunks of 4 floats) -------
__global__ void sage_scatter1_kernel(const float* __restrict__ h,
                                     const int* __restrict__ src,
                                     const int* __restrict__ dst,
                                     float* __restrict__ agg,
                                     float* __restrict__ deg) {
  unsigned tid = blockIdx.x * blockDim.x + threadIdx.x;
  unsigned e = tid >> 6;               // 256/4 = 64 chunks per edge
  if (e >= CN1) return;
  unsigned c = (tid & 63u) << 2;
  int s = src[e];
  int d = dst[e];
  const float4 v = *reinterpret_cast<const float4*>(h + (size_t)s * CD_H + c);
  float* a = agg + (size_t)d * CD_H + c;
  atomicAdd(a + 0, v.x);
  atomicAdd(a + 1, v.y);
  atomicAdd(a + 2, v.z);
  atomicAdd(a + 3, v.w);
  if ((tid & 63u) == 0u) atomicAdd(deg + d, 1.0f);
}

// -------------------- fused SAGE GEMM: one wave computes a 16x16 tile ------
// out[M,N] = act( hself[M,K] @ Wself[K,N] + (agg[M,K]/deg[M]) @ Wneigh[K,N] + bias[N] )
//
// V_WMMA_F32_16X16X4_F32 operand layout (wave32):
//   A (16x4):  lane L holds row M=L&15; VGPR0/1 = K={kh, kh+1}, kh = (L>=16)?2:0
//   B (4x16):  lane L holds col N=L&15; VGPR0/1 = K={kh, kh+1}
//   C/D 16x16: VGPR r holds row M = r + 8*(L>=16), col N = L&15
template <int K, int N, bool RELU>
__global__ void sage_gemm_kernel(const float* __restrict__ hself,
                                 const float* __restrict__ agg,
                                 const float* __restrict__ deg,
                                 const float* __restrict__ Wself,
                                 const float* __restrict__ Wneigh,
                                 const float* __restrict__ bias,
                                 float* __restrict__ out) {
  const int lane = threadIdx.x;           // 0..31, block == one wave32
  const int half = lane >> 4;             // 0 or 1
  const int l15  = lane & 15;
  const int kh   = half * 2;              // K sub-offset inside group of 4
  const int rowA = blockIdx.x * 16 + l15; // A-matrix row held by this lane
  const int col  = blockIdx.y * 16 + l15; // B/C/D column held by this lane

  const float invd = 1.0f / fmaxf(deg[rowA], 1.0f);
  const float* __restrict__ hs = hself + (size_t)rowA * K;
  const float* __restrict__ hn = agg   + (size_t)rowA * K;

  v8f acc = {};
#pragma unroll 4
  for (int k0 = 0; k0 < K; k0 += 4) {
    v2f aS, aN, bS, bN;
    aS.x = hs[k0 + kh];
    aS.y = hs[k0 + kh + 1];
    aN.x = hn[k0 + kh] * invd;
    aN.y = hn[k0 + kh + 1] * invd;
    bS.x = Wself[(size_t)(k0 + kh) * N + col];
    bS.y = Wself[(size_t)(k0 + kh + 1) * N + col];
    bN.x = Wneigh[(size_t)(k0 + kh) * N + col];
    bN.y = Wneigh[(size_t)(k0 + kh + 1) * N + col];
    // D = A*B + C  (8 args: neg_a, A, neg_b, B, c_mod, C, reuse_a, reuse_b)
    acc = __builtin_amdgcn_wmma_f32_16x16x4_f32(false, aS, false, bS,
                                                (short)0, acc, false, false);
    acc = __builtin_amdgcn_wmma_f32_16x16x4_f32(false, aN, false, bN,
                                                (short)0, acc, false, false);
  }

  const float bv = bias[col];
#pragma unroll
  for (int r = 0; r < 8; ++r) {
    const int m = blockIdx.x * 16 + r + half * 8;
    float v = acc[r] + bv;
    if (RELU) v = fmaxf(v, 0.0f);
    out[(size_t)m * N + col] = v;
  }
}

// ---------------------------------------------------------------------------
extern "C" void kernel_launch(void* const* d_in, const int* in_sizes, int n_in,
                              void* d_out, int out_size, void* d_ws, size_t ws_size,
                              hipStream_t stream) {
  (void)in_sizes; (void)n_in; (void)out_size; (void)ws_size;

  const float* x      = (const float*)d_in[0];
  const int*   src0   = (const int*)d_in[1];
  const int*   dst0   = (const int*)d_in[2];
  const int*   src1   = (const int*)d_in[3];
  const int*   dst1   = (const int*)d_in[4];
  const float* Wself0 = (const float*)d_in[5];
  const float* Wneigh0= (const float*)d_in[6];
  const float* b0     = (const float*)d_in[7];
  const float* Wself1 = (const float*)d_in[8];
  const float* Wneigh1= (const float*)d_in[9];
  const float* b1     = (const float*)d_in[10];
  float* out = (float*)d_out;

  // Workspace layout (floats):
  //   agg0: N1*128 | deg0: N1 | agg1: N2*256 | deg1: N2 | h1: N1*256
  float* ws   = (float*)d_ws;
  float* agg0 = ws;
  float* deg0 = agg0 + (size_t)CN1 * CD_IN;
  float* agg1 = deg0 + CN1;
  float* deg1 = agg1 + (size_t)CN2 * CD_H;
  float* h1   = deg1 + CN2;

  // 1) zero the accumulators (agg0|deg0|agg1|deg1 are contiguous)
  const unsigned long long nzero =
      (unsigned long long)CN1 * CD_IN + CN1 + (unsigned long long)CN2 * CD_H + CN2;
  sage_zero_kernel<<<4096, 256, 0, stream>>>(agg0, nzero);

  // 2) layer-0 scatter: N0 edges * 32 chunks
  {
    unsigned threads = (unsigned)CN0 * 32u;
    sage_scatter0_kernel<<<(threads + 255u) / 256u, 256, 0, stream>>>(
        x, src0, dst0, agg0, deg0);
  }

  // 3) layer-0 fused GEMM + ReLU -> h1  (M=81920, K=128, N=256)
  {
    dim3 grid(CN1 / 16, CD_H / 16);
    sage_gemm_kernel<CD_IN, CD_H, true><<<grid, 32, 0, stream>>>(
        x, agg0, deg0, Wself0, Wneigh0, b0, h1);
  }

  // 4) layer-1 scatter: N1 edges * 64 chunks
  {
    unsigned threads = (unsigned)CN1 * 64u;
    sage_scatter1_kernel<<<(threads + 255u) / 256u, 256, 0, stream>>>(
        h1, src1, dst1, agg1, deg1);
  }

  // 5) layer-1 fused GEMM -> out  (M=8192, K=256, N=64)
  {
    dim3 grid(CN2 / 16, CD_OUT / 16);
    sage_gemm_kernel<CD_H, CD_OUT, false><<<grid, 32, 0, stream>>>(
        h1, agg1, deg1, Wself1, Wneigh1, b1, out);
  }
}